// MVSNet_71511205478651
// MI455X (gfx1250) — compile-verified
//
#include <hip/hip_runtime.h>
#include <hip/hip_bf16.h>
#include <math.h>

// ---------------------------------------------------------------------------
// MVSNet forward for MI455X (gfx1250, wave32).
// Problem dims are fixed by setup_inputs().
// ---------------------------------------------------------------------------
namespace {
constexpr int Bn = 2, Vn = 3, Cn = 32, Hn = 128, Wn = 160, Dn = 32;
constexpr int HW = Hn * Wn;
constexpr long SIM_OFF  = 0;
constexpr long PROB_OFF = SIM_OFF  + (long)Bn * 2 * Dn * HW;   // 2,621,440
constexpr long DSP_OFF  = PROB_OFF + (long)Bn * 4 * Dn * HW;   // 7,864,320
constexpr long DVC_OFF  = DSP_OFF  + (long)Bn * 4 * HW;        // 8,028,160
constexpr long CONF_OFF = DVC_OFF  + (long)Bn * 4 * HW;        // 8,192,000
}

typedef __attribute__((ext_vector_type(2))) float v2f;
typedef __attribute__((ext_vector_type(8))) float v8f;

// Standard 4x4 inverse (adjugate / gluInvertMatrix).
__device__ inline void inv4x4(const float m[16], float o[16]) {
  float i0  =  m[5]*m[10]*m[15] - m[5]*m[11]*m[14] - m[9]*m[6]*m[15] + m[9]*m[7]*m[14] + m[13]*m[6]*m[11] - m[13]*m[7]*m[10];
  float i4  = -m[4]*m[10]*m[15] + m[4]*m[11]*m[14] + m[8]*m[6]*m[15] - m[8]*m[7]*m[14] - m[12]*m[6]*m[11] + m[12]*m[7]*m[10];
  float i8  =  m[4]*m[9]*m[15]  - m[4]*m[11]*m[13] - m[8]*m[5]*m[15] + m[8]*m[7]*m[13] + m[12]*m[5]*m[11] - m[12]*m[7]*m[9];
  float i12 = -m[4]*m[9]*m[14]  + m[4]*m[10]*m[13] + m[8]*m[5]*m[14] - m[8]*m[6]*m[13] - m[12]*m[5]*m[10] + m[12]*m[6]*m[9];
  float i1  = -m[1]*m[10]*m[15] + m[1]*m[11]*m[14] + m[9]*m[2]*m[15] - m[9]*m[3]*m[14] - m[13]*m[2]*m[11] + m[13]*m[3]*m[10];
  float i5  =  m[0]*m[10]*m[15] - m[0]*m[11]*m[14] - m[8]*m[2]*m[15] + m[8]*m[3]*m[14] + m[12]*m[2]*m[11] - m[12]*m[3]*m[10];
  float i9  = -m[0]*m[9]*m[15]  + m[0]*m[11]*m[13] + m[8]*m[1]*m[15] - m[8]*m[3]*m[13] - m[12]*m[1]*m[11] + m[12]*m[3]*m[9];
  float i13 =  m[0]*m[9]*m[14]  - m[0]*m[10]*m[13] - m[8]*m[1]*m[14] + m[8]*m[2]*m[13] + m[12]*m[1]*m[10] - m[12]*m[2]*m[9];
  float i2  =  m[1]*m[6]*m[15]  - m[1]*m[7]*m[14]  - m[5]*m[2]*m[15] + m[5]*m[3]*m[14] + m[13]*m[2]*m[7]  - m[13]*m[3]*m[6];
  float i6  = -m[0]*m[6]*m[15]  + m[0]*m[7]*m[14]  + m[4]*m[2]*m[15] - m[4]*m[3]*m[14] - m[12]*m[2]*m[7]  + m[12]*m[3]*m[6];
  float i10 =  m[0]*m[5]*m[15]  - m[0]*m[7]*m[13]  - m[4]*m[1]*m[15] + m[4]*m[3]*m[13] + m[12]*m[1]*m[7]  - m[12]*m[3]*m[5];
  float i14 = -m[0]*m[5]*m[14]  + m[0]*m[6]*m[13]  + m[4]*m[1]*m[14] - m[4]*m[2]*m[13] - m[12]*m[1]*m[6]  + m[12]*m[2]*m[5];
  float i3  = -m[1]*m[6]*m[11]  + m[1]*m[7]*m[10]  + m[5]*m[2]*m[11] - m[5]*m[3]*m[10] - m[9]*m[2]*m[7]   + m[9]*m[3]*m[6];
  float i7  =  m[0]*m[6]*m[11]  - m[0]*m[7]*m[10]  - m[4]*m[2]*m[11] + m[4]*m[3]*m[10] + m[8]*m[2]*m[7]   - m[8]*m[3]*m[6];
  float i11 = -m[0]*m[5]*m[11]  + m[0]*m[7]*m[9]   + m[4]*m[1]*m[11] - m[4]*m[3]*m[9]  - m[8]*m[1]*m[7]   + m[8]*m[3]*m[5];
  float i15 =  m[0]*m[5]*m[10]  - m[0]*m[6]*m[9]   - m[4]*m[1]*m[10] + m[4]*m[2]*m[9]  + m[8]*m[1]*m[6]   - m[8]*m[2]*m[5];
  float det = m[0]*i0 + m[1]*i4 + m[2]*i8 + m[3]*i12;
  float id  = 1.0f / det;
  o[0]=i0*id;  o[1]=i1*id;  o[2]=i2*id;  o[3]=i3*id;
  o[4]=i4*id;  o[5]=i5*id;  o[6]=i6*id;  o[7]=i7*id;
  o[8]=i8*id;  o[9]=i9*id;  o[10]=i10*id; o[11]=i11*id;
  o[12]=i12*id; o[13]=i13*id; o[14]=i14*id; o[15]=i15*id;
}

// P[b][v] = extr with top 3x4 replaced by K[:3,:3] @ E[:3,:4]. Element (r,k).
__device__ inline float Pelem(const float* __restrict__ proj, int b, int v, int r, int k) {
  const float* E = proj + ((size_t)(b * Vn + v) * 2 + 0) * 16;
  const float* K = proj + ((size_t)(b * Vn + v) * 2 + 1) * 16;
  float top = K[r*4+0]*E[0*4+k] + K[r*4+1]*E[1*4+k] + K[r*4+2]*E[2*4+k];
  return (r < 3) ? top : E[3*4+k];
}

// One wave. A(16x4): four stacked P_src matrices (pairs p=0..3 = (b, v-1)).
// B(4x16): columns 0..3 = inv(P_ref(b=0)), columns 4..7 = inv(P_ref(b=1)).
// Single V_WMMA_F32_16X16X4_F32 computes all four 4x4 products.
// ws layout per pair p: [r00..r22, t0, t1, t2] (12 floats).
__global__ void mvs_setup_matrices(const float* __restrict__ proj, float* __restrict__ ws) {
  const int lane = threadIdx.x;                 // 0..31, full wave32
  const int m    = lane & 15;                   // A row
  const int kb   = (lane < 16) ? 0 : 2;         // K pair held by this lane half
  const int pair = m >> 2;                      // 0..3 -> (b, v)
  const int ab   = pair >> 1;
  const int av   = (pair & 1) + 1;
  const int r    = m & 3;
  v2f Av;
  Av.x = Pelem(proj, ab, av, r, kb);
  Av.y = Pelem(proj, ab, av, r, kb + 1);

  // Build inv(P_ref) for the b this lane's B-column needs (redundant per lane).
  const int n    = lane & 15;                   // B column
  const int refb = (n >> 2) & 1;
  float Pref[16], Pinv[16];
  #pragma unroll
  for (int rr = 0; rr < 4; ++rr)
    #pragma unroll
    for (int kk = 0; kk < 4; ++kk)
      Pref[rr * 4 + kk] = Pelem(proj, refb, 0, rr, kk);
  inv4x4(Pref, Pinv);

  v2f Bv;
  const int col = n & 3;
  const bool live = (n < 8);
  Bv.x = live ? Pinv[kb * 4 + col]       : 0.0f;
  Bv.y = live ? Pinv[(kb + 1) * 4 + col] : 0.0f;

  v8f Cv = {};
  v8f Dv = __builtin_amdgcn_wmma_f32_16x16x4_f32(
      /*neg_a=*/false, Av, /*neg_b=*/false, Bv,
      /*c_mod=*/(short)0, Cv, /*reuse_a=*/false, /*reuse_b=*/false);

  __shared__ float Dl[16][16];
  #pragma unroll
  for (int g = 0; g < 8; ++g) {
    int mm = (lane < 16) ? g : (8 + g);
    Dl[mm][lane & 15] = Dv[g];
  }
  __syncthreads();
  if (lane < 4) {
    const int p = lane;
    const int j = p >> 1;                       // ref-b block column
    float* o = ws + p * 12;
    #pragma unroll
    for (int rr = 0; rr < 3; ++rr) {
      #pragma unroll
      for (int cc = 0; cc < 3; ++cc) o[rr * 3 + cc] = Dl[4 * p + rr][4 * j + cc];
      o[9 + rr] = Dl[4 * p + rr][4 * j + 3];
    }
  }
}

// Block = (32 depths) x (8 pixels along w). Ref features + depths staged in LDS.
__global__ __launch_bounds__(256)
void mvs_sim(const float* __restrict__ feat, const float* __restrict__ depth,
             const float* __restrict__ warpM, float* __restrict__ sim) {
  __shared__ float refS[8][32];   // [pixel][channel]
  __shared__ float depS[8][32];   // [pixel][depth]
  const int tx = threadIdx.x;     // depth index
  const int ty = threadIdx.y;     // pixel-in-block
  const int w0 = blockIdx.x * 8;
  const int h  = blockIdx.y;
  const int b  = blockIdx.z;
  const int tid = ty * 32 + tx;
  {
    int p = tid >> 5, c = tid & 31;
    refS[p][c] = feat[((size_t)(b * Vn + 0) * Cn + c) * HW + h * Wn + w0 + p];
  }
  {
    int p = tid & 7, dd = tid >> 3;
    depS[p][dd] = depth[((size_t)b * Dn + dd) * HW + h * Wn + w0 + p];
  }
  __syncthreads();

  const int w = w0 + ty;
  const float dep = depS[ty][tx];
  const float xf = (float)w, yf = (float)h;
  float acc0 = 0.0f, acc1 = 0.0f;

  #pragma unroll
  for (int v = 1; v < Vn; ++v) {
    const float* M = warpM + ((b * 2) + (v - 1)) * 12;
    const float r00 = M[0], r01 = M[1], r02 = M[2];
    const float r10 = M[3], r11 = M[4], r12 = M[5];
    const float r20 = M[6], r21 = M[7], r22 = M[8];
    const float t0 = M[9], t1 = M[10], t2 = M[11];

    const float X = (r00 * xf + r01 * yf + r02) * dep + t0;
    const float Y = (r10 * xf + r11 * yf + r12) * dep + t1;
    const float Z = (r20 * xf + r21 * yf + r22) * dep + t2;
    const float iz = 1.0f / Z;
    const float x = X * iz, y = Y * iz;

    const float x0f = floorf(x), y0f = floorf(y);
    const float fx = x - x0f, fy = y - y0f;
    const float gx = 1.0f - fx, gy = 1.0f - fy;

    const bool vx0 = (x0f >= 0.0f)  && (x0f <= (float)(Wn - 1));
    const bool vx1 = (x0f >= -1.0f) && (x0f <= (float)(Wn - 2));
    const bool vy0 = (y0f >= 0.0f)  && (y0f <= (float)(Hn - 1));
    const bool vy1 = (y0f >= -1.0f) && (y0f <= (float)(Hn - 2));
    const float w00 = (vx0 && vy0) ? gx * gy : 0.0f;
    const float w10 = (vx1 && vy0) ? fx * gy : 0.0f;
    const float w01 = (vx0 && vy1) ? gx * fy : 0.0f;
    const float w11 = (vx1 && vy1) ? fx * fy : 0.0f;

    // NaN/huge-safe clamped integer coords (weights are already zeroed OOB).
    const int x0c = (int)fminf(fmaxf(x0f,        0.0f), (float)(Wn - 1));
    const int x1c = (int)fminf(fmaxf(x0f + 1.0f, 0.0f), (float)(Wn - 1));
    const int y0c = (int)fminf(fmaxf(y0f,        0.0f), (float)(Hn - 1));
    const int y1c = (int)fminf(fmaxf(y0f + 1.0f, 0.0f), (float)(Hn - 1));

    const int o00 = y0c * Wn + x0c, o10 = y0c * Wn + x1c;
    const int o01 = y1c * Wn + x0c, o11 = y1c * Wn + x1c;
    const float* src = feat + ((size_t)(b * Vn + v) * Cn) * HW;

    #pragma unroll 4
    for (int c = 0; c < Cn; c += 2) {
      const float* p0 = src + (size_t)c * HW;
      const float* p1 = p0 + HW;
      float a0 = p0[o00] * w00 + p0[o10] * w10 + p0[o01] * w01 + p0[o11] * w11;
      float a1 = p1[o00] * w00 + p1[o10] * w10 + p1[o01] * w01 + p1[o11] * w11;
      acc0 += a0 * refS[ty][c];
      acc1 += a1 * refS[ty][c + 1];
    }
  }
  const size_t so = ((size_t)(b * 2 + 0) * Dn + tx) * HW + h * Wn + w;
  sim[so] = acc0 * (1.0f / 16.0f);
  sim[so + (size_t)Dn * HW] = acc1 * (1.0f / 16.0f);
}

// One thread per (b, q, h, w): softmax over D in registers + depth expectation.
__global__ __launch_bounds__(256)
void mvs_softmax(const float* __restrict__ cost, const float* __restrict__ depth,
                 float* __restrict__ prob, float* __restrict__ dsp) {
  const int idx = blockIdx.x * 256 + threadIdx.x;
  if (idx >= Bn * 4 * HW) return;
  const int pix = idx % HW;
  const int bq  = idx / HW;                   // b*4 + q
  const int b   = bq >> 2;

  const float* cr = cost + (size_t)bq * Dn * HW + pix;
  float vals[Dn];
  float mx = -3.402823466e38f;
  #pragma unroll
  for (int d = 0; d < Dn; ++d) { vals[d] = cr[(size_t)d * HW]; mx = fmaxf(mx, vals[d]); }
  float s = 0.0f;
  #pragma unroll
  for (int d = 0; d < Dn; ++d) { vals[d] = __expf(vals[d] - mx); s += vals[d]; }
  const float is = 1.0f / s;

  const float* dv = depth + (size_t)b * Dn * HW + pix;
  float* pr = prob + (size_t)bq * Dn * HW + pix;
  float acc = 0.0f;
  #pragma unroll
  for (int d = 0; d < Dn; ++d) {
    float p = vals[d] * is;
    pr[(size_t)d * HW] = p;
    acc += p * dv[(size_t)d * HW];
  }
  dsp[idx] = acc;
}

// One thread per (b, h, w): dvc stack6 selection + variance confidence.
__global__ __launch_bounds__(256)
void mvs_post(const float* __restrict__ dsp, const float* __restrict__ interval,
              float* __restrict__ dvc, float* __restrict__ conf) {
  const int idx = blockIdx.x * 256 + threadIdx.x;
  if (idx >= Bn * HW) return;
  const int pix = idx % HW;
  const int b   = idx / HW;
  const int h   = pix / Wn;
  const int w   = pix % Wn;

  const float d0 = dsp[(size_t)(b * 4 + 0) * HW + pix];
  const float d1 = dsp[(size_t)(b * 4 + 1) * HW + pix];
  const float d2 = dsp[(size_t)(b * 4 + 2) * HW + pix];
  const float d3 = dsp[(size_t)(b * 4 + 3) * HW + pix];

  const float smn = fminf(d0, d1), smx = fmaxf(d0, d1);
  const float hmn = fminf(d2, d3), hmx = fmaxf(d2, d3);
  const float smnd = 2.0f * smn - smx, smxd = 2.0f * smx - smn;
  const float hmnd = 2.0f * hmn - hmx, hmxd = 2.0f * hmx - hmn;

  const int r4 = h & 3, c2 = w & 1;
  float mn, mx; int off;
  if (r4 == 0)      { mn = smn;  mx = smx;  off = c2 ? 2 : 0; }
  else if (r4 == 1) { mn = hmn;  mx = hmx;  off = c2 ? 0 : 2; }
  else if (r4 == 2) { mn = smnd; mx = smxd; off = c2 ? 2 : 0; }
  else              { mn = hmnd; mx = hmxd; off = c2 ? 0 : 2; }

  const float st0 = 3.0f * mn - 2.0f * mx;
  const float st1 = 2.0f * mn - mx;
  const float st2 = mn;
  const float st3 = mx;
  const float st4 = 2.0f * mx - mn;
  const float st5 = 3.0f * mx - 2.0f * mn;
  const float e0 = (off == 0) ? st0 : st2;
  const float e1 = (off == 0) ? st1 : st3;
  const float e2 = (off == 0) ? st2 : st4;
  const float e3 = (off == 0) ? st3 : st5;
  dvc[(size_t)(b * 4 + 0) * HW + pix] = e0;
  dvc[(size_t)(b * 4 + 1) * HW + pix] = e1;
  dvc[(size_t)(b * 4 + 2) * HW + pix] = e2;
  dvc[(size_t)(b * 4 + 3) * HW + pix] = e3;

  const float mu = 0.25f * (d0 + d1 + d2 + d3);
  const float var = 0.25f * ((d0 - mu) * (d0 - mu) + (d1 - mu) * (d1 - mu) +
                             (d2 - mu) * (d2 - mu) + (d3 - mu) * (d3 - mu));
  const float z = interval[0] / (sqrtf(var) + 1e-5f);
  const float sg = 1.0f / (1.0f + __expf(-z));
  conf[idx] = 2.0f * (sg - 0.5f);
}

extern "C" void kernel_launch(void* const* d_in, const int* in_sizes, int n_in,
                              void* d_out, int out_size, void* d_ws, size_t ws_size,
                              hipStream_t stream) {
  (void)in_sizes; (void)n_in; (void)out_size; (void)ws_size;
  const float* features = (const float*)d_in[0];
  const float* proj     = (const float*)d_in[1];
  const float* depth    = (const float*)d_in[2];
  const float* cost     = (const float*)d_in[3];
  const float* interval = (const float*)d_in[4];

  float* out  = (float*)d_out;
  float* sim  = out + SIM_OFF;
  float* prob = out + PROB_OFF;
  float* dsp  = out + DSP_OFF;
  float* dvc  = out + DVC_OFF;
  float* conf = out + CONF_OFF;
  float* warpM = (float*)d_ws;   // 4 x 12 floats

  mvs_setup_matrices<<<1, 32, 0, stream>>>(proj, warpM);

  dim3 blk(32, 8);
  dim3 grd(Wn / 8, Hn, Bn);
  mvs_sim<<<grd, blk, 0, stream>>>(features, depth, warpM, sim);

  mvs_softmax<<<(Bn * 4 * HW + 255) / 256, 256, 0, stream>>>(cost, depth, prob, dsp);
  mvs_post<<<(Bn * HW + 255) / 256, 256, 0, stream>>>(dsp, interval, dvc, conf);
}